// ForwardTransformLayer_18485539242196
// MI455X (gfx1250) — compile-verified
//
#include <hip/hip_runtime.h>

// ---------------------------------------------------------------------------
// Forward DWT (8-tap circular conv + stride-2 downsample) on MI455X / gfx1250.
//
// Memory-bound problem (512 MB traffic, ~1 GFLOP) -> stream through LDS with
// coalesced loads, do the banded-matmul MACs on the native fp32 WMMA
// (v_wmma_f32_16x16x4_f32) to keep full fp32 precision at zero extra traffic.
// ---------------------------------------------------------------------------

typedef __attribute__((ext_vector_type(2))) float v2f;
typedef __attribute__((ext_vector_type(8))) float v8f;

#define N_COLS    16384      // input row length
#define N_ROWS    4096
#define OUT_COLS  8192       // per-filter output row length (stride-2)
#define TILE_U    264        // input columns staged per block (256 + 8 halo)
#define LDS_STRIDE 268       // padded LDS row stride (floats): 268%64=12,
                             // gcd(12,64)=4 -> 16 rows hit 16 distinct bank
                             // groups; even stride keeps b64 loads 8B aligned
#define BLOCK_THREADS 256    // 8 wave32 waves

__global__ __launch_bounds__(BLOCK_THREADS)
void dwt_fwd_wmma_f32(const float* __restrict__ x,
                      const float* __restrict__ s,
                      float* __restrict__ out)
{
    __shared__ float lds[16 * LDS_STRIDE];
    __shared__ float filt[8];

    const int tid  = threadIdx.x;
    const int lane = tid & 31;
    const int wave = tid >> 5;

    const int c0blk = blockIdx.x * 128;   // first output column of this block
    const int row0  = blockIdx.y * 16;    // first row of this block
    const int u0    = 2 * c0blk - 8;      // global input col of LDS col 0

    if (tid < 8) filt[tid] = s[tid];

    // ---- stage 16 x 264 input tile into LDS (coalesced; circular wrap) ----
    for (int i = tid; i < 16 * TILE_U; i += BLOCK_THREADS) {
        int r = i / TILE_U;
        int u = i - r * TILE_U;
        int g = u0 + u;                   // in [-8, N_COLS-1]; only block 0 wraps
        if (g < 0) g += N_COLS;
        lds[r * LDS_STRIDE + u] = x[(size_t)(row0 + r) * N_COLS + g];
    }
    __syncthreads();

    // ---- per-lane band-matrix B slices (tile-position independent) --------
    // B_p[k,c] = filt[2c + 8 - 4p - k], zero outside tap range [0,7].
    // Layout per ISA: lane holds N = lane%16; VGPR j holds K = j + 2*(lane/16).
    const int c  = lane & 15;             // B column / output column in tile
    const int kb = (lane >> 4) << 1;      // K base for this half-wave

    float Bh[10][2];                      // scaling filter h (approximation)
    float Bg[10][2];                      // wavelet g[j] = h[7-j]*(-1)^j (details)
#pragma unroll
    for (int p = 0; p < 10; ++p) {
#pragma unroll
        for (int q = 0; q < 2; ++q) {
            int j = 2 * c + 8 - 4 * p - (kb + q);
            float h = 0.0f, g = 0.0f;
            if (j >= 0 && j < 8) {
                h = filt[j];
                float w = filt[7 - j];
                g = (j & 1) ? -w : w;
            }
            Bh[p][q] = h;
            Bg[p][q] = g;
        }
    }

    // ---- 16x16 output tile per wave: 10 K=4 chunks x 2 filters ------------
    // A layout per ISA: lane holds M = lane%16; VGPR j holds K = j + 2*(lane/16).
    const int base = 32 * wave;           // LDS col of this wave's band start
    const int arow = lane & 15;

    v8f acc_h = {0.f, 0.f, 0.f, 0.f, 0.f, 0.f, 0.f, 0.f};
    v8f acc_g = {0.f, 0.f, 0.f, 0.f, 0.f, 0.f, 0.f, 0.f};
#pragma unroll
    for (int p = 0; p < 10; ++p) {
        // conflict-free, 8B-aligned ds_load_b64: A[r, 4p+kb .. 4p+kb+1]
        const v2f a = *(const v2f*)&lds[arow * LDS_STRIDE + base + 4 * p + kb];
        v2f bg; bg.x = Bg[p][0]; bg.y = Bg[p][1];
        v2f bh; bh.x = Bh[p][0]; bh.y = Bh[p][1];
        // D = A x B + C, fp32 in / fp32 accumulate
        acc_g = __builtin_amdgcn_wmma_f32_16x16x4_f32(
            false, a, false, bg, (short)0, acc_g, false, false);
        acc_h = __builtin_amdgcn_wmma_f32_16x16x4_f32(
            false, a, false, bh, (short)0, acc_h, false, false);
    }

    // ---- write D tiles: details first, then approximation -----------------
    // C/D layout: VGPR v -> row = v + 8*(lane/16), col = lane%16.
    const int outcol = c0blk + 16 * wave + c;
    const int rbase  = row0 + ((lane >> 4) << 3);
    float* det = out;
    float* app = out + (size_t)N_ROWS * OUT_COLS;
#pragma unroll
    for (int v = 0; v < 8; ++v) {
        size_t idx = (size_t)(rbase + v) * OUT_COLS + outcol;
        det[idx] = acc_g[v];
        app[idx] = acc_h[v];
    }
}

extern "C" void kernel_launch(void* const* d_in, const int* in_sizes, int n_in,
                              void* d_out, int out_size, void* d_ws, size_t ws_size,
                              hipStream_t stream)
{
    (void)in_sizes; (void)n_in; (void)out_size; (void)d_ws; (void)ws_size;
    const float* x = (const float*)d_in[0];   // (4096, 16384) fp32
    const float* s = (const float*)d_in[1];   // (8,) fp32
    float* out = (float*)d_out;               // details ++ approximation

    dim3 grid(OUT_COLS / 128, N_ROWS / 16);   // (64, 256)
    dim3 block(BLOCK_THREADS);
    hipLaunchKernelGGL(dwt_fwd_wmma_f32, grid, block, 0, stream, x, s, out);
}